// SelfAttention_with_mask_5738076307907
// MI455X (gfx1250) — compile-verified
//
#include <hip/hip_runtime.h>
#include <hip/hip_bf16.h>
#include <math.h>

#define Bq   2
#define Lq   512
#define HINq 1024
#define ENTq 16
#define Dq   50
#define NE1q 17           // ENT+1
#define TOTq 850          // NE1*D
#define DPAD 64
#define SCALEq 0.14142135623730950488f   // 1/sqrt(50)

typedef float v2f __attribute__((ext_vector_type(2)));
typedef float v4f __attribute__((ext_vector_type(4)));
typedef float v8f __attribute__((ext_vector_type(8)));

#define PROJ_ELEMS ((size_t)Bq * NE1q * Lq * DPAD)   // 1,114,112 floats
#define CD_ELEMS   ((size_t)Bq * (Lq - 1))

#define LDS_STRIDE 20   // 16 K-floats + 4 pad: bank-spread and 16B-aligned rows

// ---------------------------------------------------------------------------
// Kernel 1: projection GEMM  h = x @ W + bias, scattered to [b][e][l][64].
// LDS-blocked: 4-wave workgroup computes a 64x64 tile; A(64x16) and B(16x64,
// stored transposed) chunks staged in LDS; each wave owns 16 M-rows and all
// four 16-wide N subtiles (A fragment reused 4x per WMMA step).
// blockIdx.z selects start(0)/end(1) weights.
// ---------------------------------------------------------------------------
__global__ __launch_bounds__(128)
void proj_gemm_kernel(const float* __restrict__ x,
                      const float* __restrict__ Ws, const float* __restrict__ bs,
                      const float* __restrict__ We, const float* __restrict__ be,
                      float* __restrict__ projS, float* __restrict__ projE)
{
    __shared__ float As[64 * LDS_STRIDE];   // [row M][K]   5 KB
    __shared__ float Bs[64 * LDS_STRIDE];   // [col N][K]   5 KB (transposed)

    const int tid  = threadIdx.x;
    const int lane = tid & 31;
    const int w    = tid >> 5;                    // wave id 0..3 -> M subtile
    const int n0   = blockIdx.x * 64;             // N strip base (0..832)
    const int m0   = blockIdx.y * 64;             // M strip base (0..960)
    const bool isEnd = (blockIdx.z != 0);

    const float* __restrict__ W    = isEnd ? We : Ws;
    const float* __restrict__ bias = isEnd ? be : bs;
    float* __restrict__ dst        = isEnd ? projE : projS;

    const int mr = lane & 15;                     // M (A frag) / N (B frag)
    const int kh = (lane >> 4) * 2;               // K sub-offset: 0 or 2

    v8f acc[4] = {};

    for (int k0 = 0; k0 < HINq; k0 += 16) {
        // ---- stage A chunk: x[m0..m0+63][k0..k0+15] -> As[row][k] ----
        #pragma unroll
        for (int rep = 0; rep < 2; ++rep) {
            const int f   = tid + rep * 128;      // 0..255 float4 slots
            const int row = f >> 2;
            const int kk  = (f & 3) * 4;
            v4f v = *(const v4f*)(x + (size_t)(m0 + row) * HINq + k0 + kk);
            *(v4f*)&As[row * LDS_STRIDE + kk] = v;
        }
        // ---- stage B chunk transposed: W[k0..k0+15][n0..n0+63] -> Bs[n][k] ----
        #pragma unroll
        for (int rep = 0; rep < 8; ++rep) {
            const int f = tid + rep * 128;        // 0..1023
            const int k = f >> 6;
            const int n = f & 63;
            int col = n0 + n;
            if (col >= TOTq) col = TOTq - 1;      // clamp tail strip (dropped later)
            Bs[n * LDS_STRIDE + k] = W[(size_t)(k0 + k) * TOTq + col];
        }
        __syncthreads();

        // ---- 4 WMMA K-steps over the 16-deep chunk ----
        #pragma unroll
        for (int ks = 0; ks < 16; ks += 4) {
            v2f a  = *(const v2f*)&As[(w * 16 + mr) * LDS_STRIDE + ks + kh];
            v2f b0 = *(const v2f*)&Bs[( 0 + mr) * LDS_STRIDE + ks + kh];
            v2f b1 = *(const v2f*)&Bs[(16 + mr) * LDS_STRIDE + ks + kh];
            v2f b2 = *(const v2f*)&Bs[(32 + mr) * LDS_STRIDE + ks + kh];
            v2f b3 = *(const v2f*)&Bs[(48 + mr) * LDS_STRIDE + ks + kh];
            acc[0] = __builtin_amdgcn_wmma_f32_16x16x4_f32(false, a, false, b0,
                                                           (short)0, acc[0], false, false);
            acc[1] = __builtin_amdgcn_wmma_f32_16x16x4_f32(false, a, false, b1,
                                                           (short)0, acc[1], false, false);
            acc[2] = __builtin_amdgcn_wmma_f32_16x16x4_f32(false, a, false, b2,
                                                           (short)0, acc[2], false, false);
            acc[3] = __builtin_amdgcn_wmma_f32_16x16x4_f32(false, a, false, b3,
                                                           (short)0, acc[3], false, false);
        }
        __syncthreads();
    }

    // ---- epilogue: +bias, scatter into [b][e][l][64] padded layout ----
    #pragma unroll
    for (int s = 0; s < 4; ++s) {
        const int col = n0 + s * 16 + mr;
        if (col < TOTq) {
            const float bv = bias[col];
            const int e = col / Dq;
            const int d = col - e * Dq;
            #pragma unroll
            for (int r = 0; r < 8; ++r) {
                const int M   = (lane < 16) ? r : (8 + r);
                const int row = m0 + w * 16 + M;          // 0..1023
                const int bb  = row >> 9;
                const int l   = row & (Lq - 1);
                dst[(((size_t)bb * NE1q + e) * Lq + l) * DPAD + d] = acc[s][r] + bv;
            }
        }
    }
}

// ---------------------------------------------------------------------------
// Kernel 2: RoPE on entities 0..15 of both projections (in place).
// ---------------------------------------------------------------------------
__global__ void rope_kernel(float* __restrict__ projS, float* __restrict__ projE)
{
    const int total = 2 * Bq * ENTq * Lq * (Dq / 2);
    int tid = blockIdx.x * blockDim.x + threadIdx.x;
    if (tid >= total) return;
    int p = tid % (Dq / 2); int r = tid / (Dq / 2);
    int l = r % Lq;  r /= Lq;
    int e = r % ENTq; r /= ENTq;
    int b = r % Bq;   r /= Bq;
    float* base = (r ? projE : projS)
                + (((size_t)b * NE1q + e) * Lq + l) * DPAD + 2 * p;
    const float invf = __expf((float)p * (-2.0f / (float)Dq) * 9.210340371976184f);
    const float ang  = (float)l * invf;
    float s, c;
    __sincosf(ang, &s, &c);
    const float x0 = base[0], x1 = base[1];
    base[0] = x0 * c - x1 * s;
    base[1] = x1 * c + x0 * s;
}

// ---------------------------------------------------------------------------
// Kernel 3: conj_diag[b][k] = scale * dot(conj_start[b,k,:], conj_end[b,k+1,:])
// ---------------------------------------------------------------------------
__global__ void conj_diag_kernel(const float* __restrict__ projS,
                                 const float* __restrict__ projE,
                                 float* __restrict__ cd)
{
    int tid = blockIdx.x * blockDim.x + threadIdx.x;
    int b = tid / Lq, k = tid % Lq;
    if (b >= Bq || k >= Lq - 1) return;
    const float* cs = projS + (((size_t)b * NE1q + ENTq) * Lq + k) * DPAD;
    const float* ce = projE + (((size_t)b * NE1q + ENTq) * Lq + (k + 1)) * DPAD;
    float acc = 0.0f;
    #pragma unroll
    for (int d = 0; d < Dq; ++d) acc = fmaf(cs[d], ce[d], acc);
    cd[b * (Lq - 1) + k] = acc * SCALEq;
}

// ---------------------------------------------------------------------------
// Kernel 4: final_mask[b][i][j] = sum_k conj_diag[b][k] * mask[k][i][j]
// 536 MB HBM stream -> the roofline floor (~23 us @ 23.3 TB/s). float4 loads.
// ---------------------------------------------------------------------------
__global__ __launch_bounds__(256)
void final_mask_kernel(const float* __restrict__ mask,
                       const float* __restrict__ cd,
                       float* __restrict__ fm)
{
    __shared__ float s_cd[Bq * (Lq - 1)];
    for (int t = threadIdx.x; t < Bq * (Lq - 1); t += blockDim.x)
        s_cd[t] = cd[t];
    __syncthreads();

    const int idx4 = (blockIdx.x * blockDim.x + threadIdx.x) * 4;  // flat (i,j) base
    v4f a0 = {}, a1 = {};
    for (int k = 0; k < Lq - 1; ++k) {
        if (k + 8 < Lq - 1)
            __builtin_prefetch(mask + (size_t)(k + 8) * (Lq * Lq) + idx4, 0, 1);
        const v4f m = *(const v4f*)(mask + (size_t)k * (Lq * Lq) + idx4);
        a0 += s_cd[k] * m;
        a1 += s_cd[(Lq - 1) + k] * m;
    }
    *(v4f*)(fm + idx4) = a0;
    *(v4f*)(fm + (size_t)Lq * Lq + idx4) = a1;
}

// ---------------------------------------------------------------------------
// Kernel 5: scores[b,e] = rope(q) @ rope(k)^T * scale ; out = scores + fm.
// 512-thread block = 16 waves, wave id == entity e. Each wave does a 16x16
// (i,j) WMMA GEMM over K=64 (zero-padded), drops its tile into LDS [i][j][e],
// then the whole block streams the 16x(16*16) tile out fully coalesced with
// the final_mask broadcast add fused in.
// ---------------------------------------------------------------------------
__global__ __launch_bounds__(512)
void scores_kernel(const float* __restrict__ projS,
                   const float* __restrict__ projE,
                   const float* __restrict__ fm,
                   float* __restrict__ out)
{
    __shared__ float tile[16 * 16 * ENTq];        // [i][j][e] flat, 16 KB

    const int tid  = threadIdx.x;
    const int lane = tid & 31;
    const int e    = tid >> 5;                    // wave id == entity
    const int j0   = blockIdx.x * 16;
    const int i0   = blockIdx.y * 16;
    const int b    = blockIdx.z;

    const int mr = lane & 15;
    const int kh = (lane >> 4) * 2;

    const float* aptr = projS + (((size_t)b * NE1q + e) * Lq + i0 + mr) * DPAD + kh;
    const float* bptr = projE + (((size_t)b * NE1q + e) * Lq + j0 + mr) * DPAD + kh;

    v8f acc = {};
    #pragma unroll
    for (int k0 = 0; k0 < DPAD; k0 += 4) {
        v2f a    = *(const v2f*)(aptr + k0);
        v2f bmat = *(const v2f*)(bptr + k0);
        acc = __builtin_amdgcn_wmma_f32_16x16x4_f32(false, a, false, bmat,
                                                    (short)0, acc, false, false);
    }

    #pragma unroll
    for (int r = 0; r < 8; ++r) {
        const int iL = (lane < 16) ? r : (8 + r);
        tile[iL * 256 + mr * 16 + e] = acc[r] * SCALEq;
    }
    __syncthreads();

    // coalesced write-out: each thread owns 8 consecutive floats of one row
    const int iL = tid >> 5;                      // 0..15
    const int c  = (tid & 31) * 8;                // 0..248, spans one j (8|16)
    const int i  = i0 + iL;
    const int j  = j0 + (c >> 4);
    const float f = fm[((size_t)b * Lq + i) * Lq + j];

    const float* sp = &tile[iL * 256 + c];
    float* dstp = out + (((size_t)b * Lq + i) * Lq + j0) * ENTq + c;
    v4f v0 = *(const v4f*)sp       + f;
    v4f v1 = *(const v4f*)(sp + 4) + f;
    *(v4f*)dstp       = v0;
    *(v4f*)(dstp + 4) = v1;
}

// ---------------------------------------------------------------------------
extern "C" void kernel_launch(void* const* d_in, const int* in_sizes, int n_in,
                              void* d_out, int out_size, void* d_ws, size_t ws_size,
                              hipStream_t stream)
{
    const float* x    = (const float*)d_in[0];   // (B, L, HIN)
    const float* mask = (const float*)d_in[1];   // (L-1, L, L)
    const float* Ws   = (const float*)d_in[2];   // (HIN, TOT)
    const float* bs   = (const float*)d_in[3];   // (TOT,)
    const float* We   = (const float*)d_in[4];   // (HIN, TOT)
    const float* bev  = (const float*)d_in[5];   // (TOT,)
    float* out = (float*)d_out;                  // (B, L, L, ENT)

    float* ws    = (float*)d_ws;
    float* projS = ws;                           // [B][17][L][64]
    float* projE = projS + PROJ_ELEMS;
    float* cd    = projE + PROJ_ELEMS;           // [B][511]
    float* fm    = cd + CD_ELEMS;                // [B][L][L]

    // zero the D=50->64 padding so the K=64 WMMA loop in scores is exact
    hipMemsetAsync(projS, 0, 2 * PROJ_ELEMS * sizeof(float), stream);

    // 1. projections (start + end via grid.z), LDS-blocked WMMA GEMM
    proj_gemm_kernel<<<dim3((TOTq + 63) / 64, (Bq * Lq) / 64, 2), 128, 0, stream>>>(
        x, Ws, bs, We, bev, projS, projE);

    // 2. RoPE on entities 0..15
    {
        const int total = 2 * Bq * ENTq * Lq * (Dq / 2);
        rope_kernel<<<(total + 255) / 256, 256, 0, stream>>>(projS, projE);
    }

    // 3. conjunction diagonal
    conj_diag_kernel<<<(Bq * Lq + 255) / 256, 256, 0, stream>>>(projS, projE, cd);

    // 4. mask stream (bandwidth-dominant), float4
    final_mask_kernel<<<(Lq * Lq / 4) / 256, 256, 0, stream>>>(mask, cd, fm);

    // 5. attention scores + fused mask add, coalesced output
    scores_kernel<<<dim3(Lq / 16, Lq / 16, Bq), 512, 0, stream>>>(
        projS, projE, fm, out);
}